// MATLAB_SSIM_31456340476298
// MI455X (gfx1250) — compile-verified
//
#include <hip/hip_runtime.h>
#include <hip/hip_bf16.h>

typedef float v2f __attribute__((ext_vector_type(2)));
typedef float v8f __attribute__((ext_vector_type(8)));

#define IMG_N 4096

// cv2.getGaussianKernel(11, 1.5), normalized
__constant__ float GW[11] = {
    0.00102838f, 0.00759876f, 0.03600077f, 0.10936069f, 0.21300553f,
    0.26601172f, 0.21300553f, 0.10936069f, 0.03600077f, 0.00759876f,
    0.00102838f};

// One wave computes a 16(row) x 32(col) SSIM tile with fp32 WMMA.
// 8 waves / block (256 threads), tiles: 256 x 128 = 32768 waves = 4096 blocks.
__global__ __launch_bounds__(256) void ssim_wmma_kernel(
    const float* __restrict__ img1, const float* __restrict__ img2,
    float* __restrict__ partial)
{
    __shared__ float wtab[64];          // band lookup: wtab[16+i] = w[i], else 0
    __shared__ float Tlds[8][32 * 16];  // per-wave H-pass intermediate (one N-block)
    __shared__ float red[8];

    const int tid  = threadIdx.x;
    const int wv   = tid >> 5;
    const int lane = tid & 31;
    const int j    = lane & 15;   // N-col (B/C/D) or M-row (A) within 16
    const int hi   = lane >> 4;   // half-wave select (K split)

    if (tid < 64) {
        int i = tid - 16;
        wtab[tid] = (i >= 0 && i <= 10) ? GW[i] : 0.0f;
    }
    __syncthreads();

    // Banded Gaussian coefficient fragments. By symmetry of the 16x4 A-layout
    // and 4x16 B-layout lane striping, the SAME registers serve as:
    //   B (28x16, B[c][j]=w[c-j]) for the horizontal pass, and
    //   A (16x28, A[m][c]=w[c-m]) for the vertical pass.
    v2f coef[7];
#pragma unroll
    for (int ks = 0; ks < 7; ++ks) {
        int c0 = 4 * ks + 2 * hi - j + 16;
        coef[ks].x = wtab[c0];
        coef[ks].y = wtab[c0 + 1];
    }

    const int tileIdx = blockIdx.x * 8 + wv;
    const int row0 = (tileIdx >> 7) * 16;    // 256 row tiles
    const int col0 = (tileIdx & 127) * 32;   // 128 col tiles
    float* Tw = &Tlds[wv][0];

    const float C1 = 6.5025f;    // (0.01*255)^2
    const float C2 = 58.5225f;   // (0.03*255)^2
    float sum = 0.0f;

#pragma unroll
    for (int n = 0; n < 2; ++n) {            // two 16-wide N-blocks
        v8f acc[5][2];                       // H-pass accumulators: 5 ch x 2 M-blocks
#pragma unroll
        for (int ch = 0; ch < 5; ++ch)
#pragma unroll
            for (int mb = 0; mb < 2; ++mb)
                acc[ch][mb] = (v8f){0.f,0.f,0.f,0.f,0.f,0.f,0.f,0.f};

        // ---- Horizontal pass: T(32x16) = In(32x28) x Bcoef(28x16) ----
#pragma unroll
        for (int mb = 0; mb < 2; ++mb) {
            int r = row0 - 5 + 16 * mb + j;              // A row = lane%16
            r = r < 0 ? 0 : (r > IMG_N - 1 ? IMG_N - 1 : r);  // replicate pad
            const size_t rb = (size_t)r * IMG_N;
#pragma unroll
            for (int ks = 0; ks < 7; ++ks) {
                int cbase = col0 + 16 * n - 5 + 4 * ks + 2 * hi;
                int ca = cbase     < 0 ? 0 : (cbase     > IMG_N - 1 ? IMG_N - 1 : cbase);
                int cb = cbase + 1 < 0 ? 0 : (cbase + 1 > IMG_N - 1 ? IMG_N - 1 : cbase + 1);
                v2f x1, x2;
                x1.x = img1[rb + ca]; x1.y = img1[rb + cb];
                x2.x = img2[rb + ca]; x2.y = img2[rb + cb];
                v2f a2 = x1 * x1;   // img1^2
                v2f a3 = x2 * x2;   // img2^2
                v2f a4 = x1 * x2;   // img1*img2
                acc[0][mb] = __builtin_amdgcn_wmma_f32_16x16x4_f32(false, x1, false, coef[ks], (short)0, acc[0][mb], false, false);
                acc[1][mb] = __builtin_amdgcn_wmma_f32_16x16x4_f32(false, x2, false, coef[ks], (short)0, acc[1][mb], false, false);
                acc[2][mb] = __builtin_amdgcn_wmma_f32_16x16x4_f32(false, a2, false, coef[ks], (short)0, acc[2][mb], false, false);
                acc[3][mb] = __builtin_amdgcn_wmma_f32_16x16x4_f32(false, a3, false, coef[ks], (short)0, acc[3][mb], false, false);
                acc[4][mb] = __builtin_amdgcn_wmma_f32_16x16x4_f32(false, a4, false, coef[ks], (short)0, acc[4][mb], false, false);
            }
        }

        // ---- Vertical pass per channel: Out(16x16) = Acoef(16x28) x T(28x16) ----
        // D-layout -> B-layout conversion via per-wave LDS round trip (in-order DS).
        v8f outc[5];
#pragma unroll
        for (int ch = 0; ch < 5; ++ch) {
#pragma unroll
            for (int mb = 0; mb < 2; ++mb)
#pragma unroll
                for (int v = 0; v < 8; ++v) {
                    int rr = 16 * mb + 8 * hi + v;   // D: vgpr v = row v (+8 for hi half)
                    Tw[rr * 16 + j] = acc[ch][mb][v];
                }
            v8f o = (v8f){0.f,0.f,0.f,0.f,0.f,0.f,0.f,0.f};
#pragma unroll
            for (int ks = 0; ks < 7; ++ks) {
                int r0 = 4 * ks + 2 * hi;            // B: K rows split by half-wave
                v2f b;
                b.x = Tw[(r0 + 0) * 16 + j];
                b.y = Tw[(r0 + 1) * 16 + j];
                o = __builtin_amdgcn_wmma_f32_16x16x4_f32(false, coef[ks], false, b, (short)0, o, false, false);
            }
            outc[ch] = o;
        }

        // ---- SSIM map + lane-local accumulation ----
#pragma unroll
        for (int v = 0; v < 8; ++v) {
            float mu1 = outc[0][v], mu2 = outc[1][v];
            float m11 = outc[2][v], m22 = outc[3][v], m12 = outc[4][v];
            float mu1s = mu1 * mu1, mu2s = mu2 * mu2, mu12 = mu1 * mu2;
            float s1 = m11 - mu1s, s2 = m22 - mu2s, s12 = m12 - mu12;
            float num = (2.0f * mu12 + C1) * (2.0f * s12 + C2);
            float den = (mu1s + mu2s + C1) * (s1 + s2 + C2);
            sum += num / den;
        }
    }

    // wave32 reduce -> block reduce -> deterministic per-block partial
#pragma unroll
    for (int off = 16; off > 0; off >>= 1)
        sum += __shfl_xor(sum, off, 32);
    if (lane == 0) red[wv] = sum;
    __syncthreads();
    if (tid == 0) {
        float s = 0.0f;
#pragma unroll
        for (int w = 0; w < 8; ++w) s += red[w];
        partial[blockIdx.x] = s;
    }
}

__global__ __launch_bounds__(256) void ssim_finalize_kernel(
    const float* __restrict__ partial, float* __restrict__ out, int n)
{
    __shared__ double sd[256];
    double a = 0.0;
    for (int i = threadIdx.x; i < n; i += 256) a += (double)partial[i];
    sd[threadIdx.x] = a;
    __syncthreads();
    for (int off = 128; off > 0; off >>= 1) {
        if ((int)threadIdx.x < off) sd[threadIdx.x] += sd[threadIdx.x + off];
        __syncthreads();
    }
    if (threadIdx.x == 0)
        out[0] = (float)(sd[0] * (1.0 / ((double)IMG_N * (double)IMG_N)));
}

extern "C" void kernel_launch(void* const* d_in, const int* in_sizes, int n_in,
                              void* d_out, int out_size, void* d_ws, size_t ws_size,
                              hipStream_t stream) {
    const float* img1 = (const float*)d_in[0];
    const float* img2 = (const float*)d_in[1];
    float* out = (float*)d_out;
    float* partial = (float*)d_ws;   // 4096 floats = 16 KB of scratch

    const int nblocks = (IMG_N / 16) * (IMG_N / 32) / 8;  // 4096
    ssim_wmma_kernel<<<dim3(nblocks), dim3(256), 0, stream>>>(img1, img2, partial);
    ssim_finalize_kernel<<<dim3(1), dim3(256), 0, stream>>>(partial, out, nblocks);
}